// GIoULoss_74878459838529
// MI455X (gfx1250) — compile-verified
//
#include <hip/hip_runtime.h>

#ifndef __has_builtin
#define __has_builtin(x) 0
#endif

#define AS1 __attribute__((address_space(1)))
#define AS3 __attribute__((address_space(3)))

typedef int v4i __attribute__((ext_vector_type(4)));

namespace {
constexpr int kB = 16;        // batch
constexpr int kM = 4096;      // pred boxes per image
constexpr int kN = 512;       // target boxes per image
constexpr int kChunk = 32;    // pred rows per pipeline chunk
constexpr float kThr = 0.1f;  // IOU_THRESHOLD
constexpr float kEps = 1e-7f; // GIoU eps
}

extern "C" __global__ __launch_bounds__(256)
void giou_greedy_kernel(const float* __restrict__ pred,  // [B, M, 4]
                        const float* __restrict__ tgt,   // [B, N, 4]
                        float* __restrict__ per_image)   // [B]
{
    extern __shared__ float smem[];
    float* tx1 = smem;            // [N]
    float* ty1 = tx1 + kN;        // [N]
    float* tx2 = ty1 + kN;        // [N]
    float* ty2 = tx2 + kN;        // [N]
    float* tar = ty2 + kN;        // [N] target areas
    float* iouBuf = tar + kN;     // [2 * kChunk * kN] double-buffered IoU rows

    const int b    = blockIdx.x;
    const int tid  = threadIdx.x;
    const int lane = tid & 31;
    const int wave = tid >> 5;

    const float* predB = pred + (size_t)b * kM * 4;
    const float* tgtB  = tgt  + (size_t)b * kN * 4;

    // ---- Phase 0: stage target boxes (8 KB) into LDS; staging reuses iouBuf ----
    float* staging = iouBuf;
#if __has_builtin(__builtin_amdgcn_global_load_async_to_lds_b128)
    for (int j = tid; j < kN; j += 256) {
        AS1 v4i* g = (AS1 v4i*)(tgtB + j * 4);        // addrspacecast + const drop
        AS3 v4i* l = (AS3 v4i*)(staging + j * 4);     // generic->LDS addrspacecast
        __builtin_amdgcn_global_load_async_to_lds_b128(g, l, 0, 0);
    }
  #if __has_builtin(__builtin_amdgcn_s_wait_asynccnt)
    __builtin_amdgcn_s_wait_asynccnt(0);
  #else
    asm volatile("s_wait_asynccnt 0" ::: "memory");
  #endif
#else
    for (int j = tid; j < kN; j += 256) {
        float4 v = ((const float4*)tgtB)[j];
        ((float4*)staging)[j] = v;
    }
#endif
    __syncthreads();

    // AoS -> SoA + area precompute
    for (int j = tid; j < kN; j += 256) {
        float4 v = ((const float4*)staging)[j];
        tx1[j] = v.x; ty1[j] = v.y; tx2[j] = v.z; ty2[j] = v.w;
        tar[j] = (v.z - v.x) * (v.w - v.y);
    }
    __syncthreads();

    // ---- Phase 1: producer/consumer pipeline ----
    unsigned usedMask = 0;  // consumer: 16 targets per lane -> 512 bits per wave
    float lsum = 0.0f, lcnt = 0.0f;
    const int nch = kM / kChunk;

    for (int c = 0; c <= nch; ++c) {
        if (wave != 0 && c < nch) {
            // Producers: fill IoU rows for chunk c into buffer (c & 1)
            float* buf = iouBuf + (size_t)(c & 1) * kChunk * kN;
            const int rowBase = c * kChunk;
            if (tid == 32 && c + 1 < nch)
                __builtin_prefetch(predB + (size_t)(c + 1) * kChunk * 4, 0, 1);
            for (int t = tid - 32; t < kChunk * kN; t += 224) {
                const int rr = t >> 9;        // /kN
                const int cc = t & (kN - 1);
                float4 p = ((const float4*)predB)[rowBase + rr];
                float ix1 = fmaxf(p.x, tx1[cc]);
                float iy1 = fmaxf(p.y, ty1[cc]);
                float ix2 = fminf(p.z, tx2[cc]);
                float iy2 = fminf(p.w, ty2[cc]);
                float inter = fmaxf(ix2 - ix1, 0.0f) * fmaxf(iy2 - iy1, 0.0f);
                float pa = (p.z - p.x) * (p.w - p.y);
                float uni = pa + tar[cc] - inter;
                buf[rr * kN + cc] = inter / uni;  // reference pairwise IoU: no eps
            }
        }
        if (wave == 0 && c > 0) {
            // Consumer: greedy match rows of chunk c-1
            const float* buf = iouBuf + (size_t)((c - 1) & 1) * kChunk * kN;
            const int rowBase = (c - 1) * kChunk;
            for (int rr = 0; rr < kChunk; ++rr) {
                const float* row = buf + rr * kN + lane * 16;
                float4 v0 = ((const float4*)row)[0];
                float4 v1 = ((const float4*)row)[1];
                float4 v2 = ((const float4*)row)[2];
                float4 v3 = ((const float4*)row)[3];
                float vals[16];
                vals[0]=v0.x; vals[1]=v0.y; vals[2]=v0.z; vals[3]=v0.w;
                vals[4]=v1.x; vals[5]=v1.y; vals[6]=v1.z; vals[7]=v1.w;
                vals[8]=v2.x; vals[9]=v2.y; vals[10]=v2.z; vals[11]=v2.w;
                vals[12]=v3.x; vals[13]=v3.y; vals[14]=v3.z; vals[15]=v3.w;

                float best = -2.0f;      // iou >= 0, used -> -1
                int bestj = 0;
                #pragma unroll
                for (int k = 0; k < 16; ++k) {
                    float v = ((usedMask >> k) & 1u) ? -1.0f : vals[k];
                    if (v > best) { best = v; bestj = lane * 16 + k; }  // keep first index on ties
                }
                // wave32 butterfly argmax, first-index tie-break (matches jnp.argmax)
                #pragma unroll
                for (int off = 16; off > 0; off >>= 1) {
                    float ov = __shfl_xor(best, off, 32);
                    int   oj = __shfl_xor(bestj, off, 32);
                    if (ov > best || (ov == best && oj < bestj)) { best = ov; bestj = oj; }
                }
                const bool valid = best > kThr;
                if (valid && (bestj >> 4) == lane) usedMask |= 1u << (bestj & 15);
                if (valid && lane == 0) {
                    float4 p = ((const float4*)predB)[rowBase + rr];
                    float qx1 = tx1[bestj], qy1 = ty1[bestj];
                    float qx2 = tx2[bestj], qy2 = ty2[bestj];
                    float x1 = fmaxf(p.x, qx1), y1 = fmaxf(p.y, qy1);
                    float x2 = fminf(p.z, qx2), y2 = fminf(p.w, qy2);
                    float inter = fmaxf(x2 - x1, 0.0f) * fmaxf(y2 - y1, 0.0f);
                    float pa = (p.z - p.x) * (p.w - p.y);
                    float ta = (qx2 - qx1) * (qy2 - qy1);
                    float uni = pa + ta - inter;
                    float iou = inter / (uni + kEps);
                    float ex1 = fminf(p.x, qx1), ey1 = fminf(p.y, qy1);
                    float ex2 = fmaxf(p.z, qx2), ey2 = fmaxf(p.w, qy2);
                    float enc = (ex2 - ex1) * (ey2 - ey1);
                    float giou = iou - (enc - uni) / (enc + kEps);
                    lsum += 1.0f - giou;
                    lcnt += 1.0f;
                }
            }
        }
        __syncthreads();
    }

    if (tid == 0) per_image[b] = (lcnt > 0.0f) ? (lsum / lcnt) : 0.0f;
}

extern "C" __global__ __launch_bounds__(32)
void giou_finalize_kernel(const float* __restrict__ per_image, float* __restrict__ out)
{
    const int t = threadIdx.x;
    float v = (t < kB) ? per_image[t] : 0.0f;
    #pragma unroll
    for (int off = 16; off > 0; off >>= 1) v += __shfl_xor(v, off, 32);
    if (t == 0) out[0] = v / (float)kB;
}

extern "C" void kernel_launch(void* const* d_in, const int* in_sizes, int n_in,
                              void* d_out, int out_size, void* d_ws, size_t ws_size,
                              hipStream_t stream)
{
    (void)in_sizes; (void)n_in; (void)out_size; (void)ws_size;
    const float* pred = (const float*)d_in[0];   // [16, 4096, 4] f32
    const float* tgt  = (const float*)d_in[1];   // [16, 512, 4]  f32
    float* out = (float*)d_out;                  // scalar f32
    float* ws  = (float*)d_ws;                   // per-image results [16]

    const size_t smemBytes = (size_t)(5 * kN + 2 * kChunk * kN) * sizeof(float); // ~138 KB
    giou_greedy_kernel<<<kB, 256, smemBytes, stream>>>(pred, tgt, ws);
    giou_finalize_kernel<<<1, 32, 0, stream>>>(ws, out);
}